// DiffIoU_46918222741743
// MI455X (gfx1250) — compile-verified
//
#include <hip/hip_runtime.h>
#include <stdint.h>

#define DIM    100
#define NV     64
#define MAX_S  201                    // int(DIM/0.5)+1
#define MASK_ELEMS (DIM*DIM)          // 10000 floats
#define MASK_CHUNKS (MASK_ELEMS*4/16) // 2500 B128 transfers

// ---------------------------------------------------------------------------
// Shoelace-style area exactly as reference: |sum (x1-x0)*(ymax-(y0+y1)/2)|
// ---------------------------------------------------------------------------
__device__ __forceinline__ float shoelace_area(const float* __restrict__ p) {
  float ymax = -1e30f;
  #pragma unroll 8
  for (int v = 0; v < NV; ++v) ymax = fmaxf(ymax, p[2*v + 1]);
  float s = 0.0f;
  #pragma unroll 8
  for (int v = 0; v < NV; ++v) {
    int w = (v + 1) & (NV - 1);
    s += (p[2*w] - p[2*v]) * (ymax - 0.5f * (p[2*w + 1] + p[2*v + 1]));
  }
  return fabsf(s);
}

__global__ void DiffIoU_areas_kernel(const float* __restrict__ poly,
                                     const float* __restrict__ gt,
                                     float* __restrict__ pred_a,
                                     float* __restrict__ gt_a, int bs) {
  int b = blockIdx.x * blockDim.x + threadIdx.x;
  if (b >= bs) return;
  pred_a[b] = shoelace_area(poly + (size_t)b * 2 * NV);
  gt_a[b]   = shoelace_area(gt   + (size_t)b * 2 * NV);
}

// ---------------------------------------------------------------------------
// One block per (batch, mask-channel). Stages the 100x100 f32 mask slice
// (40 KB) into LDS with CDNA5 async global->LDS B128 DMA (ASYNCcnt-tracked),
// then evaluates 64 edges x {fwd,bwd} x 201 samples against the LDS-resident
// mask. Thread mapping: 256 threads = 128 (edge,dir) pairs x 2 sample-halves,
// so the sqrt/div edge setup is hoisted and prev-sample state is carried.
// Writes |sum_e sign_e*(sf_e+sb_e)/2| into int_part[b*4+m]. Deterministic.
// ---------------------------------------------------------------------------
__global__ __launch_bounds__(256) void
DiffIoU_intarea_kernel(const float* __restrict__ poly,
                       const float* __restrict__ gt_mask,
                       float* __restrict__ int_part) {
  __shared__ __align__(16) float s_mask[MASK_ELEMS];
  __shared__ float s_poly[2 * NV];
  __shared__ float s_red[256];

  const int  tid = threadIdx.x;
  const int  b   = blockIdx.x >> 2;
  const int  m   = blockIdx.x & 3;
  const bool axx = (m < 2);           // DIRECTIONS = x,x,y,y

  // ---- async DMA: gt_mask[b][m] -> LDS (2500 x b128, ASYNCcnt-tracked) ----
  {
    const uint64_t gbase    = (uint64_t)(uintptr_t)(gt_mask + ((size_t)b * 4 + m) * MASK_ELEMS);
    const unsigned lds_base = (unsigned)(uintptr_t)(&s_mask[0]);  // addr[31:0] == LDS offset
    #pragma unroll
    for (int it = 0; it < 10; ++it) {
      int c = tid + it * 256;
      if (c < MASK_CHUNKS) {
        unsigned goff = (unsigned)(c * 16);
        unsigned loff = lds_base + (unsigned)(c * 16);
        asm volatile("global_load_async_to_lds_b128 %0, %1, %2"
                     :: "v"(loff), "v"(goff), "s"(gbase)
                     : "memory");
      }
    }
  }
  if (tid < 2 * NV) s_poly[tid] = poly[(size_t)b * 2 * NV + tid];
  asm volatile("s_wait_asynccnt 0" ::: "memory");
  __syncthreads();

  // ---- per-thread work: one (edge, direction) pair, ~100 samples ----
  const int pair = tid >> 1;          // 0..127
  const int half = tid & 1;
  const int e    = pair >> 1;         // 0..63
  const int dir  = pair & 1;          // 0 = forward, 1 = backward
  const int s_begin = half ? 101 : 0;
  const int s_end   = half ? MAX_S : 101;

  const int e2 = (e + 1) & (NV - 1);
  const float px = s_poly[2*e],  py = s_poly[2*e + 1];
  const float qx = s_poly[2*e2], qy = s_poly[2*e2 + 1];

  const float sign = ((axx ? (qx > px) : (qy > py)) ? 1.0f : -1.0f);

  const float x0 = dir ? qx : px,  y0 = dir ? qy : py;
  const float x1 = dir ? px : qx,  y1 = dir ? py : qy;

  float vx = x1 - x0 + 1e-6f;
  float vy = y1 - y0 + 1e-6f;
  const float n = sqrtf(vx * vx + vy * vy);
  vx /= n; vy /= n;

  const float xlo = fminf(x0, x1) - 0.001f, xhi = fmaxf(x0, x1) + 0.001f;
  const float ylo = fminf(y0, y1) - 0.001f, yhi = fmaxf(y0, y1) + 0.001f;

  // evaluate (valid, floor(u)) at sample index s
  auto eval = [&](int s, bool& valid_o, float& fu_o) {
    float fs = (float)s;
    float xs = x0 + fs * vx;
    float ys = y0 + fs * vy;
    bool seg = (xs <= xhi) && (xs >= xlo) && (ys <= yhi) && (ys >= ylo);
    float u  = axx ? xs : ys;
    valid_o = seg && (u <= (float)(DIM - 1)) && (u >= 0.0f);
    fu_o    = floorf(u);
  };

  // carried prev-sample state (reference's prev_valid / prev_fu, pad = 0/false)
  bool  pvalid = false;
  float pfu    = 0.0f;
  if (s_begin > 0) eval(s_begin - 1, pvalid, pfu);

  float acc = 0.0f;
  for (int ssi = s_begin; ssi < s_end; ++ssi) {
    float fs = (float)ssi;
    float xs = x0 + fs * vx;
    float ys = y0 + fs * vy;
    bool seg = (xs <= xhi) && (xs >= xlo) && (ys <= yhi) && (ys >= ylo);
    float u  = axx ? xs : ys;
    bool valid = seg && (u <= (float)(DIM - 1)) && (u >= 0.0f);
    float fu   = floorf(u);

    bool keep = valid && (!pvalid || (fu != pfu));
    if (keep) {
      float Xf = floorf(xs), Yf = floorf(ys);
      float wx0 = Xf + 1.0f - xs;   // (X0f+1-xs), reference form
      float wx1 = xs - Xf;
      float wy0 = Yf + 1.0f - ys;
      float wy1 = ys - Yf;
      int Xi0 = (int)fminf(fmaxf(Xf,        0.0f), (float)(DIM - 1));
      int Xi1 = (int)fminf(fmaxf(Xf + 1.0f, 0.0f), (float)(DIM - 1));
      int Yi0 = (int)fminf(fmaxf(Yf,        0.0f), (float)(DIM - 1));
      int Yi1 = (int)fminf(fmaxf(Yf + 1.0f, 0.0f), (float)(DIM - 1));
      float v00 = s_mask[Yi0 * DIM + Xi0];
      float v01 = s_mask[Yi1 * DIM + Xi0];
      float v10 = s_mask[Yi0 * DIM + Xi1];
      float v11 = s_mask[Yi1 * DIM + Xi1];
      float val = wx0 * wy0 * v00 + wx0 * wy1 * v01 +
                  wx1 * wy0 * v10 + wx1 * wy1 * v11;
      acc += sign * 0.5f * val;
    }
    pvalid = valid;
    pfu    = fu;
  }

  // ---- deterministic block reduction (fixed tree order) ----
  s_red[tid] = acc;
  __syncthreads();
  for (int off = 128; off > 0; off >>= 1) {
    if (tid < off) s_red[tid] += s_red[tid + off];
    __syncthreads();
  }
  if (tid == 0) int_part[blockIdx.x] = fabsf(s_red[0]);
}

__global__ void DiffIoU_finalize_kernel(const float* __restrict__ pred_a,
                                        const float* __restrict__ gt_a,
                                        const float* __restrict__ int_part,
                                        float* __restrict__ out, int bs) {
  int b = blockIdx.x * blockDim.x + threadIdx.x;
  if (b >= bs) return;
  float ia = 0.25f * (int_part[4*b] + int_part[4*b + 1] +
                      int_part[4*b + 2] + int_part[4*b + 3]);
  out[b] = ia / (pred_a[b] + gt_a[b] - ia);
}

extern "C" void kernel_launch(void* const* d_in, const int* in_sizes, int n_in,
                              void* d_out, int out_size, void* d_ws, size_t ws_size,
                              hipStream_t stream) {
  const float* poly = (const float*)d_in[0];   // (bs, 64, 2) f32
  const float* gt   = (const float*)d_in[1];   // (bs, 64, 2) f32
  const float* gtm  = (const float*)d_in[2];   // (bs, 4, 100, 100) f32
  float* out = (float*)d_out;                  // (bs,) f32

  const int bs = in_sizes[0] / (2 * NV);

  float* pred = (float*)d_ws;                  // bs floats
  float* gta  = pred + bs;                     // bs floats
  float* ints = gta + bs;                      // bs*4 floats

  DiffIoU_areas_kernel<<<(bs + 255) / 256, 256, 0, stream>>>(poly, gt, pred, gta, bs);
  DiffIoU_intarea_kernel<<<bs * 4, 256, 0, stream>>>(poly, gtm, ints);
  DiffIoU_finalize_kernel<<<(bs + 255) / 256, 256, 0, stream>>>(pred, gta, ints, out, bs);
}